// SparseDropout_16638703304888
// MI455X (gfx1250) — compile-verified
//
#include <hip/hip_runtime.h>
#include <stdint.h>

// Sparse dropout (COO): out = (indices passthrough, mask ? values*2 : 0).
// Memory-bound: ~1.31 GB total HBM traffic -> ~56 us floor at 23.3 TB/s.
// Strategy: every memory instruction is a B128 access; non-temporal hints
// everywhere (touch-once streaming, keep L2 clean); bulk indices copy goes
// through the CDNA5 async global<->LDS engine, 4 chunks in flight per wave.
// The per-chunk stride is folded into the async ops' 24-bit IOFFSET, which
// the ISA applies to BOTH the LDS and global addresses, so the hot path has
// no per-chunk address math or predication.

typedef float    v4f __attribute__((ext_vector_type(4)));
typedef uint32_t v4u __attribute__((ext_vector_type(4)));

#define SCALE 2.0f  // 1 / kprob, kprob = 0.5
#define COPY_CHUNKS 4
#define BLOCK 256
static_assert(BLOCK * 16 == 4096, "inline-asm offsets assume 4096B chunk stride");

// ---------------------------------------------------------------------------
// Kernel 1: bulk copy of the indices bytes through the async LDS path.
// Thread t of a full block moves bytes {base + k*4096 + t*16, k=0..3} via LDS
// slots {k*4096 + t*16}: one addr VGPR pair + one LDS-offset VGPR, chunk
// stride carried entirely in the instruction offset field.
// ---------------------------------------------------------------------------
__global__ void copy_bytes_async_b128(const uint8_t* __restrict__ src,
                                      uint8_t* __restrict__ dst,
                                      long long nchunks) {
  __shared__ __align__(16) uint8_t lbuf[BLOCK * 16 * COPY_CHUNKS];
  const unsigned tid = threadIdx.x;
  const long long base = (long long)blockIdx.x * (BLOCK * COPY_CHUNKS);

  // Low 32 bits of a generic pointer into __shared__ == LDS byte offset.
  const unsigned lds_base = (unsigned)(unsigned long long)(uintptr_t)(&lbuf[0]);
  const unsigned lds_off  = lds_base + tid * 16u;

  if (base + (long long)(BLOCK * COPY_CHUNKS) <= nchunks) {
    // Fast path: full block, no predication, stride via IOFFSET.
    unsigned long long saddr =
        (unsigned long long)(uintptr_t)(src + (base + tid) * 16);
    unsigned long long daddr =
        (unsigned long long)(uintptr_t)(dst + (base + tid) * 16);

    asm volatile(
        "global_load_async_to_lds_b128 %0, %1, off th:TH_LOAD_NT\n\t"
        "global_load_async_to_lds_b128 %0, %1, off offset:4096 th:TH_LOAD_NT\n\t"
        "global_load_async_to_lds_b128 %0, %1, off offset:8192 th:TH_LOAD_NT\n\t"
        "global_load_async_to_lds_b128 %0, %1, off offset:12288 th:TH_LOAD_NT"
        :: "v"(lds_off), "v"(saddr) : "memory");

    asm volatile("s_wait_asynccnt 0" ::: "memory");

    asm volatile(
        "global_store_async_from_lds_b128 %0, %1, off th:TH_STORE_NT\n\t"
        "global_store_async_from_lds_b128 %0, %1, off offset:4096 th:TH_STORE_NT\n\t"
        "global_store_async_from_lds_b128 %0, %1, off offset:8192 th:TH_STORE_NT\n\t"
        "global_store_async_from_lds_b128 %0, %1, off offset:12288 th:TH_STORE_NT"
        :: "v"(daddr), "v"(lds_off) : "memory");
  } else {
    // Ragged last block: per-chunk predication.
#pragma unroll
    for (int k = 0; k < COPY_CHUNKS; ++k) {
      long long idx = base + (long long)k * BLOCK + tid;
      if (idx < nchunks) {
        unsigned loff = lds_base + (unsigned)(k * BLOCK + tid) * 16u;
        unsigned long long saddr = (unsigned long long)(uintptr_t)(src + idx * 16);
        asm volatile("global_load_async_to_lds_b128 %0, %1, off th:TH_LOAD_NT"
                     :: "v"(loff), "v"(saddr) : "memory");
      }
    }
    asm volatile("s_wait_asynccnt 0" ::: "memory");
#pragma unroll
    for (int k = 0; k < COPY_CHUNKS; ++k) {
      long long idx = base + (long long)k * BLOCK + tid;
      if (idx < nchunks) {
        unsigned loff = lds_base + (unsigned)(k * BLOCK + tid) * 16u;
        unsigned long long daddr = (unsigned long long)(uintptr_t)(dst + idx * 16);
        asm volatile("global_store_async_from_lds_b128 %0, %1, off th:TH_STORE_NT"
                     :: "v"(daddr), "v"(loff) : "memory");
      }
    }
  }
  // s_endpgm performs an implicit wait-idle; stores will complete.
}

// Byte tail for the copy (not hit for shape (2,N) int64, but stay generic).
__global__ void copy_bytes_tail(const uint8_t* __restrict__ src,
                                uint8_t* __restrict__ dst,
                                long long start, long long nbytes) {
  long long i = start + (long long)blockIdx.x * blockDim.x + threadIdx.x;
  if (i < nbytes) dst[i] = src[i];
}

// ---------------------------------------------------------------------------
// Kernel 2: dropout on values. 16 floats per thread: 4x B128 NT value loads,
// 1x B128 NT mask load (16 bool bytes), VALU select, 4x B128 NT stores.
// ---------------------------------------------------------------------------
__global__ void dropout_vec16(const float* __restrict__ vals,
                              const uint8_t* __restrict__ mask,
                              float* __restrict__ out,
                              long long n16) {
  long long tid = (long long)blockIdx.x * blockDim.x + threadIdx.x;
  if (tid >= n16) return;

  const v4f* vp = (const v4f*)vals + tid * 4;
  v4f v0 = __builtin_nontemporal_load(vp);
  v4f v1 = __builtin_nontemporal_load(vp + 1);
  v4f v2 = __builtin_nontemporal_load(vp + 2);
  v4f v3 = __builtin_nontemporal_load(vp + 3);

  v4u m = __builtin_nontemporal_load((const v4u*)mask + tid);

  v4f r0, r1, r2, r3;
  r0.x = (m.x & 0x000000ffu) ? v0.x * SCALE : 0.0f;
  r0.y = (m.x & 0x0000ff00u) ? v0.y * SCALE : 0.0f;
  r0.z = (m.x & 0x00ff0000u) ? v0.z * SCALE : 0.0f;
  r0.w = (m.x & 0xff000000u) ? v0.w * SCALE : 0.0f;
  r1.x = (m.y & 0x000000ffu) ? v1.x * SCALE : 0.0f;
  r1.y = (m.y & 0x0000ff00u) ? v1.y * SCALE : 0.0f;
  r1.z = (m.y & 0x00ff0000u) ? v1.z * SCALE : 0.0f;
  r1.w = (m.y & 0xff000000u) ? v1.w * SCALE : 0.0f;
  r2.x = (m.z & 0x000000ffu) ? v2.x * SCALE : 0.0f;
  r2.y = (m.z & 0x0000ff00u) ? v2.y * SCALE : 0.0f;
  r2.z = (m.z & 0x00ff0000u) ? v2.z * SCALE : 0.0f;
  r2.w = (m.z & 0xff000000u) ? v2.w * SCALE : 0.0f;
  r3.x = (m.w & 0x000000ffu) ? v3.x * SCALE : 0.0f;
  r3.y = (m.w & 0x0000ff00u) ? v3.y * SCALE : 0.0f;
  r3.z = (m.w & 0x00ff0000u) ? v3.z * SCALE : 0.0f;
  r3.w = (m.w & 0xff000000u) ? v3.w * SCALE : 0.0f;

  v4f* op = (v4f*)out + tid * 4;
  __builtin_nontemporal_store(r0, op);
  __builtin_nontemporal_store(r1, op + 1);
  __builtin_nontemporal_store(r2, op + 2);
  __builtin_nontemporal_store(r3, op + 3);
}

__global__ void dropout_tail(const float* __restrict__ vals,
                             const uint8_t* __restrict__ mask,
                             float* __restrict__ out,
                             long long start, long long n) {
  long long i = start + (long long)blockIdx.x * blockDim.x + threadIdx.x;
  if (i < n) out[i] = mask[i] ? vals[i] * SCALE : 0.0f;
}

// ---------------------------------------------------------------------------
extern "C" void kernel_launch(void* const* d_in, const int* in_sizes, int n_in,
                              void* d_out, int out_size, void* d_ws, size_t ws_size,
                              hipStream_t stream) {
  (void)n_in; (void)out_size; (void)d_ws; (void)ws_size;

  const uint8_t* idx_bytes_in = (const uint8_t*)d_in[0];  // int64 indices, raw bytes
  const float*   vals         = (const float*)d_in[1];    // float32 values
  const uint8_t* mask         = (const uint8_t*)d_in[2];  // jnp bool_ = 1 byte

  long long idx_elems = in_sizes[0];  // 2 * NNZ int64 elements
  long long nnz       = in_sizes[1];

  uint8_t* out_bytes = (uint8_t*)d_out;
  uint8_t* out_idx   = out_bytes;                         // indices passthrough
  float*   out_vals  = (float*)(out_bytes + idx_elems * 8);

  // --- indices copy via async LDS path (4 x 16B chunks per thread) ---
  long long idx_nbytes = idx_elems * 8;
  long long nchunks    = idx_nbytes / 16;
  if (nchunks > 0) {
    long long per_block = (long long)BLOCK * COPY_CHUNKS;
    long long grid = (nchunks + per_block - 1) / per_block;
    copy_bytes_async_b128<<<dim3((uint32_t)grid), dim3(BLOCK), 0, stream>>>(
        idx_bytes_in, out_idx, nchunks);
  }
  long long rem_bytes = idx_nbytes - nchunks * 16;
  if (rem_bytes > 0) {
    copy_bytes_tail<<<dim3(1), dim3(BLOCK), 0, stream>>>(
        idx_bytes_in, out_idx, nchunks * 16, idx_nbytes);
  }

  // --- dropout on values (16 floats per thread) ---
  long long n16 = nnz / 16;
  if (n16 > 0) {
    long long grid = (n16 + BLOCK - 1) / BLOCK;
    dropout_vec16<<<dim3((uint32_t)grid), dim3(BLOCK), 0, stream>>>(
        vals, mask, out_vals, n16);
  }
  long long rem = nnz - n16 * 16;
  if (rem > 0) {
    dropout_tail<<<dim3(1), dim3(BLOCK), 0, stream>>>(
        vals, mask, out_vals, n16 * 16, nnz);
  }
}